// SpringAndMassAction_84859963835011
// MI455X (gfx1250) — compile-verified
//
#include <hip/hip_runtime.h>
#include <hip/hip_bf16.h>
#include <math.h>

// ---------------------------------------------------------------------------
// SpringAndMassAction on MI455X (gfx1250, wave32, WMMA + async global->LDS)
//   Phase 1: x = softplus(kx[...,:E]) @ Wx^T + bx ; k = relu(kx[...,E:2E]) @ Wk^T + bk
//            -> fp32 WMMA 16x16x4 GEMM, results to workspace (row-major (B*N) x E)
//   Phase 2: fused Gram matrices (x x^T, k k^T), cosine cutoff, weighted P-sum,
//            tanh norm limiting. i-tiles staged to LDS with double-buffered
//            GLOBAL_LOAD_ASYNC_TO_LDS_B128 (ASYNCcnt), overlapping WMMA compute.
// ---------------------------------------------------------------------------

#define E       128
#define NN      1024
#define BB      32
#define LDSS    132   // LDS row stride (floats): 128 + 4 pad -> conflict-free frag reads
#define TI      64    // i-tile rows (double buffered)

// Async global->LDS path (gfx1250): guarded so a missing builtin degrades to the
// proven synchronous float4 staging instead of failing the build.
#if defined(__has_builtin)
#  if __has_builtin(__builtin_amdgcn_global_load_async_to_lds_b128) && \
      __has_builtin(__builtin_amdgcn_s_wait_asynccnt)
#    define SMA_ASYNC 1
#  endif
#endif
#ifndef SMA_ASYNC
#  define SMA_ASYNC 0
#endif

typedef float v2f __attribute__((ext_vector_type(2)));
typedef float v8f __attribute__((ext_vector_type(8)));
#if SMA_ASYNC
// Signature (from clang diagnostics): arg0 = AS1 int4*, arg1 = AS3 int4*.
typedef __attribute__((__vector_size__(4 * sizeof(int)))) int sma_i4;
typedef __attribute__((address_space(1))) sma_i4* gas_i4_ptr;
typedef __attribute__((address_space(3))) sma_i4* lds_i4_ptr;

__device__ __forceinline__ void sma_async_cp16(const float* g, float* l) {
    __builtin_amdgcn_global_load_async_to_lds_b128(
        (gas_i4_ptr)g, (lds_i4_ptr)l, /*offset=*/0, /*cpol=*/0);
}
#endif

__device__ __forceinline__ v8f wmma_f32(v2f a, v2f b, v8f c) {
    // D(16x16 f32) = A(16x4 f32) * B(4x16 f32) + C
    return __builtin_amdgcn_wmma_f32_16x16x4_f32(
        /*neg_a=*/false, a, /*neg_b=*/false, b,
        /*c_mod=*/(short)0, c, /*reuse_a=*/false, /*reuse_b=*/false);
}

// ---------------------------------------------------------------------------
// Phase 1: activation + dense E x E projection for both x and k.
// grid = (B*N)/128 blocks of 256 threads (8 waves). Wave w -> 16 output cols.
// ---------------------------------------------------------------------------
__global__ __launch_bounds__(256)
void sma_phase1_features(const float* __restrict__ kx,
                         const float* __restrict__ amask,
                         const float* __restrict__ Wx, const float* __restrict__ bx,
                         const float* __restrict__ Wk, const float* __restrict__ bk,
                         float* __restrict__ x_ws, float* __restrict__ k_ws)
{
    __shared__ float a_lds[128 * LDSS];

    const int tid  = threadIdx.x;
    const int lane = tid & 31;
    const int wv   = tid >> 5;        // 0..7
    const int m    = lane & 15;       // row-in-subtile / column selector
    const int kh   = lane >> 4;       // 0/1: K-half of the fragment
    const int row0 = blockIdx.x * 128;

    for (int sel = 0; sel < 2; ++sel) {
        const float* W    = sel ? Wk : Wx;
        const float* bias = sel ? bk : bx;
        float*       outp = sel ? k_ws : x_ws;

        // Stage activation(kx * mask) tile: 128 rows x 128 cols, coalesced.
        for (int idx = tid; idx < 128 * E; idx += 256) {
            const int  r = idx >> 7, c = idx & 127;
            const long g = (long)(row0 + r);
            float v = kx[g * (2 * E + 1) + sel * E + c] * amask[g];
            float a = sel ? fmaxf(v, 0.0f)
                          : ((v > 20.0f) ? v : log1pf(__expf(v)));  // softplus
            a_lds[r * LDSS + c] = a;
        }
        __syncthreads();

        // B fragments: B[kc, n] = W[ecol, kc] (out = act @ W^T). Register resident.
        const int ecol = wv * 16 + m;
        v2f bw[32];
        #pragma unroll
        for (int s = 0; s < 32; ++s)
            bw[s] = *(const v2f*)(W + (long)ecol * E + ((kh << 1) + (s << 2)));
        const float bv = bias[ecol];

        for (int ms = 0; ms < 8; ++ms) {
            v8f acc = {};
            #pragma unroll
            for (int s = 0; s < 32; ++s) {
                const int c = (kh << 1) + (s << 2);
                v2f a = *(const v2f*)(&a_lds[(ms * 16 + m) * LDSS + c]);
                acc = wmma_f32(a, bw[s], acc);
            }
            // C layout: vgpr r, lane l -> M = r + (l/16)*8, N = l%16
            #pragma unroll
            for (int r = 0; r < 8; ++r) {
                const long orow = row0 + ms * 16 + r + kh * 8;
                outp[orow * E + ecol] = acc[r] + bv;
            }
        }
        __syncthreads();
    }
}

// ---------------------------------------------------------------------------
// Phase 2: fused pairwise interaction + reduction over i.
// grid = B * (N/128): block (b, j-tile of 128). 8 waves x 16 j-columns each.
// i-tiles of 64 rows, double buffered, staged with async global->LDS.
// ---------------------------------------------------------------------------
__global__ __launch_bounds__(256)
void sma_phase2_actions(const float* __restrict__ x_ws,
                        const float* __restrict__ k_ws,
                        const float* __restrict__ positions,
                        const float* __restrict__ amask,
                        float* __restrict__ out)
{
    __shared__ float xi_lds[2][TI * LDSS];
    __shared__ float ki_lds[2][TI * LDSS];
    __shared__ float pos_lds[2][TI * 3];
    __shared__ float msk_lds[2][TI];

    const int tid  = threadIdx.x;
    const int lane = tid & 31;
    const int wv   = tid >> 5;
    const int m    = lane & 15;
    const int kh   = lane >> 4;
    const int b    = blockIdx.x >> 3;
    const int jt   = blockIdx.x & 7;
    const int j    = jt * 128 + wv * 16 + m;   // this lane's output column
    const long bN  = (long)b * NN;

    // Register-resident B fragments for this wave's 16 j-columns (x and k).
    v2f bxf[32], bkf[32];
    {
        const float* xr = x_ws + (bN + j) * E;
        const float* kr = k_ws + (bN + j) * E;
        #pragma unroll
        for (int s = 0; s < 32; ++s) {
            const int c = (kh << 1) + (s << 2);
            bxf[s] = *(const v2f*)(xr + c);
            bkf[s] = *(const v2f*)(kr + c);
        }
    }
    const float pjx = positions[(bN + j) * 3 + 0];
    const float pjy = positions[(bN + j) * 3 + 1];
    const float pjz = positions[(bN + j) * 3 + 2];
    const float mj  = amask[bN + j];

    const float inv_sqrt_e  = 0.08838834764831845f;   // 1/sqrt(128)
    const float pi_over_cut = 0.6283185307179586f;    // pi / CUTOFF

    float ax = 0.f, ay = 0.f, az = 0.f;

    // ---- tile staging: issue loads for tile `it` into buffer `bf` (no wait) ----
    auto stage = [&](int bf, int it) {
        const float* xsrc = x_ws + (bN + it * TI) * E;
        const float* ksrc = k_ws + (bN + it * TI) * E;
        for (int idx = tid; idx < TI * E / 4; idx += 256) {
            const int r = idx >> 5, c4 = (idx & 31) << 2;
#if SMA_ASYNC
            sma_async_cp16(xsrc + (long)r * E + c4, &xi_lds[bf][r * LDSS + c4]);
            sma_async_cp16(ksrc + (long)r * E + c4, &ki_lds[bf][r * LDSS + c4]);
#else
            *(float4*)(&xi_lds[bf][r * LDSS + c4]) = *(const float4*)(xsrc + (long)r * E + c4);
            *(float4*)(&ki_lds[bf][r * LDSS + c4]) = *(const float4*)(ksrc + (long)r * E + c4);
#endif
        }
        if (tid < TI) {
            const long gi = bN + it * TI + tid;
            pos_lds[bf][tid * 3 + 0] = positions[gi * 3 + 0];
            pos_lds[bf][tid * 3 + 1] = positions[gi * 3 + 1];
            pos_lds[bf][tid * 3 + 2] = positions[gi * 3 + 2];
            msk_lds[bf][tid]         = amask[gi];
        }
    };
    auto wait_async = [&]() {
#if SMA_ASYNC
        __builtin_amdgcn_s_wait_asynccnt(0);
#endif
    };

    const int NT = NN / TI;                 // 16 i-tiles
    stage(0, 0);
    wait_async();
    __syncthreads();

    for (int it = 0; it < NT; ++it) {
        const int cur = it & 1;
        if (it + 1 < NT) stage(cur ^ 1, it + 1);   // overlap next stage with compute

        for (int is = 0; is < TI / 16; ++is) {
            v8f cx = {}, ck = {};
            #pragma unroll
            for (int s = 0; s < 32; ++s) {
                const int c = (kh << 1) + (s << 2);
                v2f aX = *(const v2f*)(&xi_lds[cur][(is * 16 + m) * LDSS + c]);
                v2f aK = *(const v2f*)(&ki_lds[cur][(is * 16 + m) * LDSS + c]);
                cx = wmma_f32(aX, bxf[s], cx);
                ck = wmma_f32(aK, bkf[s], ck);
            }
            // Elementwise interaction + accumulate P * w over this lane's 8 i's.
            #pragma unroll
            for (int r = 0; r < 8; ++r) {
                const int il = is * 16 + r + kh * 8;            // local i index
                const float dx = pjx - pos_lds[cur][il * 3 + 0]; // P[i,j] = p_j - p_i
                const float dy = pjy - pos_lds[cur][il * 3 + 1];
                const float dz = pjz - pos_lds[cur][il * 3 + 2];
                const float rij = sqrtf(dx * dx + dy * dy + dz * dz);
                const float xs  = (rij - cx[r] * inv_sqrt_e) * 0.5f;
                const float ks  = ck[r] * inv_sqrt_e;
                const float fct = (rij < 5.0f)
                                    ? 0.5f * (__cosf(rij * pi_over_cut) + 1.0f)
                                    : 0.0f;
                const float w = xs * ks * fct * msk_lds[cur][il] * mj;
                ax += dx * w; ay += dy * w; az += dz * w;
            }
        }
        // Tile it+1's async loads must land (and everyone finish reading `cur`)
        // before the next iteration computes / restages.
        wait_async();
        __syncthreads();
    }

    // Lanes l and l^16 own the same j; combine their i-halves.
    ax += __shfl_xor(ax, 16, 32);
    ay += __shfl_xor(ay, 16, 32);
    az += __shfl_xor(az, 16, 32);

    if (kh == 0) {
        const float an = sqrtf(ax * ax + ay * ay + az * az) + 1e-8f;
        const float sc = tanhf(an) / an * mj;   // ACTION_SCALE = 1
        const long  o  = (bN + j) * 3;
        out[o + 0] = ax * sc;
        out[o + 1] = ay * sc;
        out[o + 2] = az * sc;
    }
}

// ---------------------------------------------------------------------------
extern "C" void kernel_launch(void* const* d_in, const int* in_sizes, int n_in,
                              void* d_out, int out_size, void* d_ws, size_t ws_size,
                              hipStream_t stream)
{
    const float* kx        = (const float*)d_in[0];
    const float* positions = (const float*)d_in[1];
    const float* amask     = (const float*)d_in[2];
    const float* Wx        = (const float*)d_in[3];
    const float* bx        = (const float*)d_in[4];
    const float* Wk        = (const float*)d_in[5];
    const float* bk        = (const float*)d_in[6];
    float* out             = (float*)d_out;

    float* x_ws = (float*)d_ws;                         // (B*N) x E
    float* k_ws = x_ws + (size_t)BB * NN * E;           // (B*N) x E  (needs 32 MB ws)

    sma_phase1_features<<<dim3(BB * NN / 128), dim3(256), 0, stream>>>(
        kx, amask, Wx, bx, Wk, bk, x_ws, k_ws);

    sma_phase2_actions<<<dim3(BB * (NN / 128)), dim3(256), 0, stream>>>(
        x_ws, k_ws, positions, amask, out);
}